// MorseBatchedLSTMStack_6906307412394
// MI455X (gfx1250) — compile-verified
//
#include <hip/hip_runtime.h>

#define T_LEN 262144
#define HID 8
#define GATES 32
#define CHUNK 64
#define NCHUNK (T_LEN / CHUNK)

typedef __attribute__((ext_vector_type(2))) float v2f;
typedef __attribute__((ext_vector_type(8))) float v8f;

// ds_swizzle_b32 group-of-32: src_lane = ((lane & and) | or) ^ xor
// imm = {xor[14:10], or[9:5], and[4:0]}
#define SWZ(v, imm) __int_as_float(__builtin_amdgcn_ds_swizzle(__float_as_int(v), (imm)))

// sigmoid(x) = 1/(1+exp(-x)); tanh(x) = 2*sigmoid(2x)-1 -> one v_exp + one v_rcp each
__device__ __forceinline__ float fast_sig(float x) {
    return __builtin_amdgcn_rcpf(1.0f + __expf(-x));
}
__device__ __forceinline__ float fast_tanh(float x) {
    return fmaf(2.0f, fast_sig(2.0f * x), -1.0f);
}

// One LSTM step for one lane-per-gate wave. g arrives as xp[t] contribution,
// w[] is this lane's W_hh row, hj/cj are this lane's h_{lane&7}, c_{lane&7}.
__device__ __forceinline__ void lstm_step(float g, const float* __restrict__ w,
                                          bool isg, float& hj, float& cj) {
    // recurrent matvec: broadcast h_k from lane k (BCAST swizzles, one round trip)
    g = fmaf(w[0], SWZ(hj, 0 << 5), g);
    g = fmaf(w[1], SWZ(hj, 1 << 5), g);
    g = fmaf(w[2], SWZ(hj, 2 << 5), g);
    g = fmaf(w[3], SWZ(hj, 3 << 5), g);
    g = fmaf(w[4], SWZ(hj, 4 << 5), g);
    g = fmaf(w[5], SWZ(hj, 5 << 5), g);
    g = fmaf(w[6], SWZ(hj, 6 << 5), g);
    g = fmaf(w[7], SWZ(hj, 7 << 5), g);
    // activation: lanes 16..23 (g-gate) tanh, others sigmoid
    const float s = fast_sig(isg ? 2.0f * g : g);
    const float a = isg ? fmaf(2.0f, s, -1.0f) : s;
    // gather i,f,g,o for this lane's hidden index j = lane&7 (one round trip)
    const float ig = SWZ(a, 0x0007);            // lane -> lane&7
    const float fg = SWZ(a, (8  << 5) | 7);     // lane -> (lane&7)|8
    const float gg = SWZ(a, (16 << 5) | 7);     // lane -> (lane&7)|16
    const float og = SWZ(a, (24 << 5) | 7);     // lane -> (lane&7)|24
    cj = fmaf(fg, cj, ig * gg);
    hj = og * fast_tanh(cj);
}

// 4-wave pipelined 2-layer LSTM (one workgroup, everything in 20KB LDS):
//   wave0 : layer-1 sequential scan, writes h1 chunk to LDS
//   wave2 : WMMA f32 16x16x4, xp2[:,0:16]  = h1_chunk @ W_ih2[0:16]^T  + bias
//   wave3 : WMMA f32 16x16x4, xp2[:,16:32] = h1_chunk @ W_ih2[16:32]^T + bias
//   wave1 : layer-2 sequential scan consuming xp2 chunk, then linear head
__global__ void __launch_bounds__(128, 1)
lstm2_fused_kernel(const float* __restrict__ x,
                   const float* __restrict__ W_ih1, const float* __restrict__ W_hh1,
                   const float* __restrict__ b_ih1, const float* __restrict__ b_hh1,
                   const float* __restrict__ W_ih2, const float* __restrict__ W_hh2,
                   const float* __restrict__ b_ih2, const float* __restrict__ b_hh2,
                   const float* __restrict__ W_lin, const float* __restrict__ b_lin,
                   float* __restrict__ out)
{
    __shared__ float h1buf[2][CHUNK * HID];     //  4 KB : layer-1 hidden chunks
    __shared__ float xpbuf[2][CHUNK * GATES];   // 16 KB : layer-2 gate projections

    // wave id is wave-uniform: force it scalar so all role branches are s_cbranch
    const int wid  = __builtin_amdgcn_readfirstlane((int)threadIdx.x) >> 5;
    const int lane = (int)threadIdx.x & 31;
    const bool isg = (lane >= 16) && (lane < 24);

    // -------- per-role persistent state --------
    float w[8];                 // recurrent weight row for this lane's gate
    float wih = 0.0f, bias = 0.0f;
    float hj = 0.0f, cj = 0.0f; // this lane's h_{lane&7}, c_{lane&7}
    float xpend = 0.0f;         // software-pipelined x[t]
    v2f  bfrag0, bfrag1;        // WMMA B fragments (loop-invariant)
    float bsplat = 0.0f;
    const int m  = lane & 15;
    const int hi = lane >> 4;
    const int k0 = hi * 2;

    if (wid == 0) {
        #pragma unroll
        for (int k2 = 0; k2 < 8; k2++) w[k2] = W_hh1[lane * 8 + k2];
        wih   = W_ih1[lane];                      // IN == 1
        bias  = b_ih1[lane] + b_hh1[lane];
        xpend = x[0];
    } else if (wid == 1) {
        #pragma unroll
        for (int k2 = 0; k2 < 8; k2++) w[k2] = W_hh2[lane * 8 + k2];
    } else {
        // B layout (f32 4x16): VGPR v holds row K=v (lanes 0-15), K=v+2 (lanes 16-31)
        const int nt = wid - 2;                   // scalar: this wave's N-tile
        const int n  = nt * 16 + m;               // gate/output column
        bfrag0.x = W_ih2[n * 8 + k0];     bfrag0.y = W_ih2[n * 8 + k0 + 1];
        bfrag1.x = W_ih2[n * 8 + 4 + k0]; bfrag1.y = W_ih2[n * 8 + 4 + k0 + 1];
        bsplat   = b_ih2[n] + b_hh2[n];
    }

    // -------- chunk pipeline: wave0 @ k, wmma @ k-1, wave1 @ k-2 --------
    for (int k = 0; k < NCHUNK + 2; k++) {
        if (wid == 0 && k < NCHUNK) {
            float* hb = h1buf[k & 1];
            const int t0 = k * CHUNK;
            for (int ib = 0; ib < CHUNK; ib += 16) {
                if (t0 + ib + 256 < T_LEN)
                    __builtin_prefetch(&x[t0 + ib + 256], 0, 0);  // global_prefetch_b8
                #pragma unroll
                for (int i2 = 0; i2 < 16; i2++) {
                    const int i = ib + i2;
                    const int t = t0 + i;
                    const float xt = xpend;                       // loaded last step
                    xpend = (t + 1 < T_LEN) ? x[t + 1] : 0.0f;    // issue next load early
                    lstm_step(fmaf(xt, wih, bias), w, isg, hj, cj);
                    if (lane < 8) hb[i * 8 + lane] = hj;
                }
            }
        }

        if (wid >= 2 && k >= 1 && k <= NCHUNK) {
            const int kc = k - 1;
            const float* hb = h1buf[kc & 1];
            float*       xb = xpbuf[kc & 1];
            const int nt = wid - 2;                               // scalar
            #pragma unroll
            for (int tt = 0; tt < 4; tt++) {                      // 4 time-tiles of 16
                // A (f32 16x4): lanes 0-15 K in {0,1}, lanes 16-31 K in {2,3}
                const float* hrow = &hb[(tt * 16 + m) * 8];
                v2f a0; a0.x = hrow[k0];     a0.y = hrow[k0 + 1];
                v2f a1; a1.x = hrow[4 + k0]; a1.y = hrow[4 + k0 + 1];
                v8f d;
                #pragma unroll
                for (int v = 0; v < 8; v++) d[v] = bsplat;        // bias depends only on N
                d = __builtin_amdgcn_wmma_f32_16x16x4_f32(
                        false, a0, false, bfrag0, (short)0, d, false, false);
                d = __builtin_amdgcn_wmma_f32_16x16x4_f32(
                        false, a1, false, bfrag1, (short)0, d, false, false);
                // D: VGPR v -> (M=v lanes 0-15 | M=v+8 lanes 16-31), N = lane&15
                #pragma unroll
                for (int v = 0; v < 8; v++)
                    xb[(tt * 16 + v + hi * 8) * 32 + nt * 16 + m] = d[v];
            }
        }

        if (wid == 1 && k >= 2) {
            const int kc = k - 2;
            const float* xb = xpbuf[kc & 1];
            #pragma unroll 8
            for (int i = 0; i < CHUNK; i++)
                lstm_step(xb[i * 32 + lane], w, isg, hj, cj);      // xp2 incl. biases
        }

        __syncthreads();
    }

    // -------- linear head on the final h2 (wave1 owns it) --------
    if (wid == 1) {
        float hk[8];
        hk[0] = SWZ(hj, 0 << 5); hk[1] = SWZ(hj, 1 << 5);
        hk[2] = SWZ(hj, 2 << 5); hk[3] = SWZ(hj, 3 << 5);
        hk[4] = SWZ(hj, 4 << 5); hk[5] = SWZ(hj, 5 << 5);
        hk[6] = SWZ(hj, 6 << 5); hk[7] = SWZ(hj, 7 << 5);
        if (lane < 6) {
            float acc = b_lin[lane];
            #pragma unroll
            for (int kk = 0; kk < 8; kk++)
                acc = fmaf(W_lin[lane * 8 + kk], hk[kk], acc);
            out[lane] = acc;
        }
    }
}

extern "C" void kernel_launch(void* const* d_in, const int* in_sizes, int n_in,
                              void* d_out, int out_size, void* d_ws, size_t ws_size,
                              hipStream_t stream) {
    (void)in_sizes; (void)n_in; (void)d_ws; (void)ws_size; (void)out_size;
    const float* x     = (const float*)d_in[0];
    const float* W_ih1 = (const float*)d_in[1];
    const float* W_hh1 = (const float*)d_in[2];
    const float* b_ih1 = (const float*)d_in[3];
    const float* b_hh1 = (const float*)d_in[4];
    const float* W_ih2 = (const float*)d_in[5];
    const float* W_hh2 = (const float*)d_in[6];
    const float* b_ih2 = (const float*)d_in[7];
    const float* b_hh2 = (const float*)d_in[8];
    const float* W_lin = (const float*)d_in[9];
    const float* b_lin = (const float*)d_in[10];
    float* out = (float*)d_out;

    lstm2_fused_kernel<<<dim3(1), dim3(128), 0, stream>>>(
        x, W_ih1, W_hh1, b_ih1, b_hh1,
        W_ih2, W_hh2, b_ih2, b_hh2, W_lin, b_lin, out);
}